// GIN_43593918054564
// MI455X (gfx1250) — compile-verified
//
#include <hip/hip_runtime.h>
#include <hip/hip_bf16.h>

#define N_NODES_C 100000
#define D_C 64
#define BN_EPS_C 1e-5f

typedef __attribute__((ext_vector_type(2))) float v2f;
typedef __attribute__((ext_vector_type(8))) float v8f;

// ---------------------------------------------------------------------------
// Kernel 1: agg = x (fold the "+x_i" of GIN into the accumulator init),
//           and zero the BN stats buffer (128 floats).
// ---------------------------------------------------------------------------
__global__ __launch_bounds__(256) void gin_init_kernel(
    const float* __restrict__ x, float* __restrict__ agg,
    float* __restrict__ stats, long long n4) {
  long long t = (long long)blockIdx.x * blockDim.x + threadIdx.x;
  if (t < n4) {
    ((float4*)agg)[t] = ((const float4*)x)[t];
  }
  if (blockIdx.x == 0 && threadIdx.x < 128) {
    stats[threadIdx.x] = 0.0f;
  }
}

// ---------------------------------------------------------------------------
// Kernel 2: edge scatter-add. 16 threads per edge; each thread moves a
// float4 (4 channels) via 4 global f32 atomic adds (no-return -> STOREcnt).
// ---------------------------------------------------------------------------
__global__ __launch_bounds__(256) void gin_scatter_kernel(
    const float* __restrict__ x,
    const long long* __restrict__ src_idx,
    const long long* __restrict__ dst_idx,
    float* __restrict__ agg, long long n_edges) {
  long long t = (long long)blockIdx.x * blockDim.x + threadIdx.x;
  long long e = t >> 4;
  int part = (int)(t & 15);
  if (e >= n_edges) return;

  long long s = src_idx[e];
  long long d = dst_idx[e];

  // Hint the next edge's source row into cache (emits global_prefetch_b8).
  if (e + 1 < n_edges) {
    long long s2 = src_idx[e + 1];
    __builtin_prefetch(x + s2 * D_C + part * 4, 0, 1);
  }

  float4 v = ((const float4*)(x + s * D_C))[part];
  float* dp = agg + d * D_C + part * 4;
  atomicAdd(dp + 0, v.x);
  atomicAdd(dp + 1, v.y);
  atomicAdd(dp + 2, v.z);
  atomicAdd(dp + 3, v.w);
}

// ---------------------------------------------------------------------------
// Kernel 3: h = agg @ W + b via V_WMMA_F32_16X16X4_F32.
// Block = 256 threads = 8 waves; W (64x64 = 16 KB) staged in LDS once.
// Each wave owns one 16-row tile, computes all four 16-col tiles,
// K=64 swept as 16 WMMA steps of K=4 (64 wmma ops per wave).
//
// ISA fragment layouts (wave32):
//   A 16x4 f32 : lanes 0-15 hold rows M=0..15 with K={k,k+1} in v0/v1,
//                lanes 16-31 hold the same rows with K={k+2,k+3}.
//   B 4x16 f32 : v0 = rows K=k (lanes 0-15) / K=k+2 (lanes 16-31),
//                v1 = rows K=k+1 / K=k+3, N = lane%16.
//   D 16x16    : VGPR r -> M=r (lanes 0-15), M=r+8 (lanes 16-31), N=lane%16.
// EXEC must be all-1s for WMMA: tail rows use clamped loads, guarded stores.
// ---------------------------------------------------------------------------
__global__ __launch_bounds__(256) void gin_gemm_wmma_kernel(
    const float* __restrict__ A,    // agg  [N_NODES, 64]
    const float* __restrict__ W,    // [64, 64] row-major
    const float* __restrict__ bias, // [64]
    float* __restrict__ H) {        // out  [N_NODES, 64]
  __shared__ float Wlds[D_C * D_C];

  int tid = threadIdx.x;
  // Cooperative load of W into LDS (4096 floats / 256 threads, float4).
  for (int i = tid; i < (D_C * D_C) / 4; i += 256) {
    ((float4*)Wlds)[i] = ((const float4*)W)[i];
  }
  __syncthreads();

  int wave   = tid >> 5;
  int lane   = tid & 31;
  int laneLo = lane & 15;
  int laneHi = lane >> 4;  // 0 or 1: which K-half of the quad this lane holds

  long long row0 = ((long long)blockIdx.x * 8 + wave) * 16;
  long long arow = row0 + laneLo;
  if (arow >= N_NODES_C) arow = N_NODES_C - 1;  // clamp: keep EXEC full
  const float* aBase = A + arow * D_C;

  v8f acc[4] = {};

  #pragma unroll
  for (int k0 = 0; k0 < D_C; k0 += 4) {
    int ka = k0 + (laneHi << 1);
    v2f a;
    a.x = aBase[ka];      // contiguous pair -> global_load_b64
    a.y = aBase[ka + 1];
    #pragma unroll
    for (int n = 0; n < 4; ++n) {
      int col = n * 16 + laneLo;
      v2f b;
      b.x = Wlds[ka * D_C + col];
      b.y = Wlds[(ka + 1) * D_C + col];
      acc[n] = __builtin_amdgcn_wmma_f32_16x16x4_f32(
          /*neg_a=*/false, a, /*neg_b=*/false, b,
          /*c_mod=*/(short)0, acc[n], /*reuse_a=*/false, /*reuse_b=*/false);
    }
  }

  #pragma unroll
  for (int n = 0; n < 4; ++n) {
    int col = n * 16 + laneLo;
    float bv = bias[col];
    #pragma unroll
    for (int r = 0; r < 8; ++r) {
      long long row = row0 + r + (laneHi ? 8 : 0);
      if (row < N_NODES_C) {
        H[row * D_C + col] = acc[n][r] + bv;
      }
    }
  }
}

// ---------------------------------------------------------------------------
// Kernel 4: per-channel sum / sum-of-squares. 256 threads = 4 row-groups x
// 64 channels; LDS tree within the block, then 128 global f32 atomics.
// ---------------------------------------------------------------------------
__global__ __launch_bounds__(256) void gin_bnstats_kernel(
    const float* __restrict__ H, float* __restrict__ stats) {
  __shared__ float redS[256];
  __shared__ float redQ[256];

  int tid = threadIdx.x;
  int ch  = tid & 63;
  int grp = tid >> 6;  // 0..3

  float s = 0.0f, q = 0.0f;
  for (long long row = (long long)blockIdx.x * 4 + grp; row < N_NODES_C;
       row += (long long)gridDim.x * 4) {
    float v = H[row * D_C + ch];
    s += v;
    q += v * v;
  }
  redS[tid] = s;
  redQ[tid] = q;
  __syncthreads();

  if (grp == 0) {
    s = redS[ch] + redS[64 + ch] + redS[128 + ch] + redS[192 + ch];
    q = redQ[ch] + redQ[64 + ch] + redQ[128 + ch] + redQ[192 + ch];
    atomicAdd(&stats[ch], s);
    atomicAdd(&stats[64 + ch], q);
  }
}

// ---------------------------------------------------------------------------
// Kernel 5: in-place BatchNorm (batch stats) + affine + ReLU on d_out.
// ---------------------------------------------------------------------------
__global__ __launch_bounds__(256) void gin_bnapply_kernel(
    float* __restrict__ H, const float* __restrict__ stats,
    const float* __restrict__ gamma, const float* __restrict__ beta) {
  long long t = (long long)blockIdx.x * blockDim.x + threadIdx.x;
  if (t >= (long long)N_NODES_C * D_C) return;
  int ch = (int)(t & 63);
  const float invN = 1.0f / (float)N_NODES_C;
  float mu  = stats[ch] * invN;
  float var = stats[64 + ch] * invN - mu * mu;
  float inv = rsqrtf(var + BN_EPS_C);
  float v = (H[t] - mu) * inv * gamma[ch] + beta[ch];
  H[t] = v > 0.0f ? v : 0.0f;
}

// ---------------------------------------------------------------------------
// Launch
// ---------------------------------------------------------------------------
extern "C" void kernel_launch(void* const* d_in, const int* in_sizes, int n_in,
                              void* d_out, int out_size, void* d_ws, size_t ws_size,
                              hipStream_t stream) {
  const float*     x     = (const float*)d_in[0];
  const long long* eidx  = (const long long*)d_in[1];  // int64, shape (2, E)
  const float*     W     = (const float*)d_in[2];
  const float*     bias  = (const float*)d_in[3];
  const float*     gamma = (const float*)d_in[4];
  const float*     beta  = (const float*)d_in[5];
  float*           out   = (float*)d_out;

  long long n_edges = (long long)(in_sizes[1] / 2);
  const long long* src = eidx;
  const long long* dst = eidx + n_edges;

  // Workspace layout: agg [N*64 f32] | stats [128 f32]
  float* agg   = (float*)d_ws;
  float* stats = agg + (long long)N_NODES_C * D_C;

  // 1) agg = x ; stats = 0
  long long n4 = (long long)N_NODES_C * D_C / 4;  // 1.6M float4
  int g1 = (int)((n4 + 255) / 256);
  gin_init_kernel<<<g1, 256, 0, stream>>>(x, agg, stats, n4);

  // 2) scatter-add edges
  long long sthreads = n_edges * 16;
  int g2 = (int)((sthreads + 255) / 256);
  gin_scatter_kernel<<<g2, 256, 0, stream>>>(x, src, dst, agg, n_edges);

  // 3) h = agg @ W + b  (WMMA) -> d_out
  int rowTiles = (N_NODES_C + 15) / 16;     // 6250
  int g3 = (rowTiles + 7) / 8;              // 8 waves (tiles) per block
  gin_gemm_wmma_kernel<<<g3, 256, 0, stream>>>(agg, W, bias, out);

  // 4) BN stats
  gin_bnstats_kernel<<<256, 256, 0, stream>>>(out, stats);

  // 5) BN apply + ReLU, in place on d_out
  long long elems = (long long)N_NODES_C * D_C;
  int g5 = (int)((elems + 255) / 256);
  gin_bnapply_kernel<<<g5, 256, 0, stream>>>(out, stats, gamma, beta);
}